// VSC3x3Rulebook_50354196578890
// MI455X (gfx1250) — compile-verified
//
#include <hip/hip_runtime.h>

// ---------------------------------------------------------------------------
// VSC 3x3 rulebook sparse conv for MI455X (gfx1250, wave32).
//
// D = A(16x4 f32) x B(4x16 f32) + C via V_WMMA_F32_16X16X4_F32.
//   A layout (per ISA): lane l (l<16): M=l, VGPR0=K0,VGPR1=K1 ; lanes 16-31:
//                       M=l-16, VGPR0=K2,VGPR1=K3.
//   B layout (mirror):  VGPR0: lanes0-15 K=0 N=lane, lanes16-31 K=2 ;
//                       VGPR1: K=1 / K=3.
//   C/D layout:         VGPRv: lanes0-15 M=v, lanes16-31 M=v+8, N=lane&15.
// ---------------------------------------------------------------------------

typedef __attribute__((ext_vector_type(2))) float v2f;
typedef __attribute__((ext_vector_type(8))) float v8f;

#define C_IN   64
#define C_OUT  64
#define WAVES   8
#define TPW     4          // M-tiles per wave
#define TILE_M 16
#define ENTRIES_PER_WG (WAVES * TPW * TILE_M)   // 512

// ---------------------------------------------------------------------------
// Init: out[n][c] = bias[c]   (also erases the 0xAA poison)
// ---------------------------------------------------------------------------
__global__ void vsc_init_kernel(float* __restrict__ outF,
                                const float* __restrict__ bias,
                                long long total) {
    long long stride = (long long)gridDim.x * blockDim.x;
    for (long long i = (long long)blockIdx.x * blockDim.x + threadIdx.x;
         i < total; i += stride) {
        outF[i] = bias[i & (C_OUT - 1)];
    }
}

// ---------------------------------------------------------------------------
// Main: gather -> WMMA f32 GEMM -> atomic scatter-add
// ---------------------------------------------------------------------------
__global__ __launch_bounds__(256) void vsc_gemm_kernel(
    const float* __restrict__ feats,     // [N, 64]
    const int*   __restrict__ in_rows,   // [9, M]
    const int*   __restrict__ out_rows,  // [9, M]
    const float* __restrict__ weight,    // [9, 64, 64]
    float*       __restrict__ out,       // [N, 64]
    int M, int blocksPerK)
{
    // Weight tile for this offset, pre-swizzled into WMMA-B fragment order:
    // ldsW[frag*64 + v*32 + lane] = W[4*kk + v + 2*(lane>>4)][n*16 + (lane&15)]
    // with frag = kk*4 + n.  One conflict-free ds_load_b64 per fragment.
    __shared__ float ldsW[C_IN * C_OUT];

    const int k    = blockIdx.x / blocksPerK;
    const int wg   = blockIdx.x % blocksPerK;
    const int tid  = threadIdx.x;
    const int lane = tid & 31;
    const int wave = tid >> 5;
    const int hh   = lane >> 4;    // half-wave: 0 or 1
    const int lm   = lane & 15;

    // ---- cooperative swizzled preload of weight[k] (64x64 f32, 16 KB) ----
    const float* Wk = weight + (size_t)k * C_IN * C_OUT;
    for (int d = tid; d < C_IN * C_OUT; d += blockDim.x) {
        int l    = d & 31;
        int v    = (d >> 5) & 1;
        int n    = (d >> 6) & 3;
        int kk   = d >> 8;
        int krow = 4 * kk + v + 2 * (l >> 4);
        int col  = n * 16 + (l & 15);
        ldsW[d] = Wk[krow * C_OUT + col];
    }
    __syncthreads();

    // ---- hoist all 64 B fragments (128 VGPRs) into registers ----
    v2f barr[64];
#pragma unroll
    for (int f = 0; f < 64; ++f) {
        barr[f] = *(const v2f*)&ldsW[f * 64 + 2 * lane];
    }

    const long long kbase = (long long)k * M;

    for (int t = 0; t < TPW; ++t) {
        const int m0 = wg * ENTRIES_PER_WG + (wave * TPW + t) * TILE_M;
        if (m0 >= M) break;   // wave-uniform: EXEC stays all-ones for WMMA

        // Gathered A row for this lane (two lanes per row, even/odd K pairs)
        const int rin = in_rows[kbase + m0 + lm];
        const float* aPtr = feats + (size_t)rin * C_IN + 2 * hh;

        v8f acc0 = {}, acc1 = {}, acc2 = {}, acc3 = {};

#pragma unroll
        for (int kk = 0; kk < 16; ++kk) {
            v2f a = *(const v2f*)(aPtr + 4 * kk);   // global_load_b64
            acc0 = __builtin_amdgcn_wmma_f32_16x16x4_f32(
                false, a, false, barr[kk * 4 + 0], (short)0, acc0, false, false);
            acc1 = __builtin_amdgcn_wmma_f32_16x16x4_f32(
                false, a, false, barr[kk * 4 + 1], (short)0, acc1, false, false);
            acc2 = __builtin_amdgcn_wmma_f32_16x16x4_f32(
                false, a, false, barr[kk * 4 + 2], (short)0, acc2, false, false);
            acc3 = __builtin_amdgcn_wmma_f32_16x16x4_f32(
                false, a, false, barr[kk * 4 + 3], (short)0, acc3, false, false);
        }

        // ---- scatter-add: lanes 0-15 of each atomic hit 64 contiguous B ----
#pragma unroll
        for (int v = 0; v < 8; ++v) {
            const int orow = out_rows[kbase + m0 + v + 8 * hh];
            float* op = out + (size_t)orow * C_OUT + lm;
            atomicAdd(op +  0, acc0[v]);
            atomicAdd(op + 16, acc1[v]);
            atomicAdd(op + 32, acc2[v]);
            atomicAdd(op + 48, acc3[v]);
        }
    }
}

// ---------------------------------------------------------------------------
extern "C" void kernel_launch(void* const* d_in, const int* in_sizes, int n_in,
                              void* d_out, int out_size, void* d_ws, size_t ws_size,
                              hipStream_t stream) {
    const void*  coords   = d_in[0];                 // [N,3] int32 (passthrough)
    const float* feats    = (const float*)d_in[1];   // [N,64] f32
    const int*   in_rows  = (const int*)  d_in[2];   // [9,M]
    const int*   out_rows = (const int*)  d_in[3];   // [9,M]
    const float* weight   = (const float*)d_in[4];   // [9,64,64]
    const float* bias     = (const float*)d_in[5];   // [64]

    const int N = in_sizes[0] / 3;
    const int M = in_sizes[2] / 9;

    float* outCoords = (float*)d_out;                // int32 bits pass through
    float* outFeat   = outCoords + (size_t)N * 3;

    // coords passthrough (tuple output, concatenated flat)
    hipMemcpyAsync(outCoords, coords, (size_t)N * 3 * sizeof(int),
                   hipMemcpyDeviceToDevice, stream);

    // out = bias (broadcast); overwrites poison
    const long long total = (long long)N * C_OUT;
    vsc_init_kernel<<<4096, 256, 0, stream>>>(outFeat, bias, total);

    const int blocksPerK = (M + ENTRIES_PER_WG - 1) / ENTRIES_PER_WG;
    vsc_gemm_kernel<<<dim3(9 * blocksPerK), 256, 0, stream>>>(
        feats, in_rows, out_rows, weight, outFeat, M, blocksPerK);
}